// Net_RNN_11390253269731
// MI455X (gfx1250) — compile-verified
//
#include <hip/hip_runtime.h>
#include <math.h>

#define FDIM 20
#define MPAD 32   // padded stride for the message buffer (kills epilogue divergence)

typedef __attribute__((ext_vector_type(2))) float v2f;
typedef __attribute__((ext_vector_type(8))) float v8f;

// ---------------------------------------------------------------------------
// Degree / normalization
// ---------------------------------------------------------------------------
__global__ void k_deg_init(float* __restrict__ deg, int n) {
    int i = blockIdx.x * blockDim.x + threadIdx.x;
    if (i < n) deg[i] = 1.0f;                       // self-loop contribution
}

__global__ void k_deg_scatter(const int* __restrict__ dst, float* __restrict__ deg, int e) {
    int i = blockIdx.x * blockDim.x + threadIdx.x;
    if (i < e) atomicAdd(&deg[dst[i]], 1.0f);
}

__global__ void k_rsqrt_inplace(float* __restrict__ deg, int n) {
    int i = blockIdx.x * blockDim.x + threadIdx.x;
    if (i < n) deg[i] = rsqrtf(deg[i]);             // deg >= 1 always (self loops)
}

// ---------------------------------------------------------------------------
// fc1: h = relu(x @ W + b),  x:[n,2]  W:[2,20]
// ---------------------------------------------------------------------------
__global__ void k_fc1(const float* __restrict__ x, const float* __restrict__ w,
                      const float* __restrict__ b, float* __restrict__ h, int n) {
    int idx = blockIdx.x * blockDim.x + threadIdx.x;
    if (idx >= n * FDIM) return;
    int i = idx / FDIM;
    int j = idx - i * FDIM;
    float v = fmaf(x[i * 2 + 0], w[j], fmaf(x[i * 2 + 1], w[FDIM + j], b[j]));
    h[idx] = fmaxf(v, 0.0f);
}

// ---------------------------------------------------------------------------
// msg[n,32] = (h @ W) * dinv[:,None]   via V_WMMA_F32_16X16X4_F32
// One wave computes a 16-node x 32-col (20 real) output tile: 2 col tiles,
// 5 K-steps each. Cols 20..31 of B are zero -> those outputs are exact zeros,
// so the full-tile path stores all 32 columns unconditionally (no divergence).
// EXEC is all-1s at every WMMA (guards are clamped-index selects).
// ---------------------------------------------------------------------------
__global__ void k_gemm_scale(const float* __restrict__ h,   // [n, 20]
                             const float* __restrict__ W,   // [20, 20]
                             const float* __restrict__ dinv,
                             float* __restrict__ out,       // [n, 32] padded
                             int n) {
    int wave = (int)((blockIdx.x * (unsigned)blockDim.x + threadIdx.x) >> 5);
    int lane = threadIdx.x & 31;
    int m0 = wave * 16;
    if (m0 >= n) return;                            // wave-uniform exit only

    int mrow = m0 + (lane & 15);
    int mclamp = (mrow < n) ? mrow : (n - 1);       // clamp (tail tiles), no branch
    int khalf = (lane >> 4) << 1;                   // lanes 0-15 -> K {0,1}; 16-31 -> {2,3}
    int ncol = lane & 15;
    bool fulltile = (m0 + 16 <= n);                 // wave-uniform
    int rbase = m0 + ((lane >> 4) << 3);            // lanes 0-15 rows m0+r; 16-31 m0+8+r

#pragma unroll
    for (int c0 = 0; c0 < MPAD; c0 += 16) {         // column tiles 0..15, 16..31
        int col = c0 + ncol;
        int colc = (col < FDIM) ? col : (FDIM - 1);
        v8f acc = {0.f, 0.f, 0.f, 0.f, 0.f, 0.f, 0.f, 0.f};
#pragma unroll
        for (int kk = 0; kk < FDIM; kk += 4) {      // K = 20 -> 5 steps of 4
            int ka = kk + khalf;
            v2f a, b;
            a[0] = h[mclamp * FDIM + ka];
            a[1] = h[mclamp * FDIM + ka + 1];
            float b0 = W[ka * FDIM + colc];
            float b1 = W[(ka + 1) * FDIM + colc];
            b[0] = (col < FDIM) ? b0 : 0.0f;        // v_cndmask, EXEC untouched
            b[1] = (col < FDIM) ? b1 : 0.0f;
            acc = __builtin_amdgcn_wmma_f32_16x16x4_f32(
                false, a, false, b, (short)0, acc, false, false);
        }
        if (fulltile) {                             // fast path: no guards at all
            float dv[8];
#pragma unroll
            for (int r = 0; r < 8; ++r) dv[r] = dinv[rbase + r];
#pragma unroll
            for (int r = 0; r < 8; ++r)
                out[(rbase + r) * MPAD + col] = acc[r] * dv[r];
        } else {                                    // tail (never hit for n%16==0)
#pragma unroll
            for (int r = 0; r < 8; ++r) {
                int row = rbase + r;
                if (row < n)
                    out[row * MPAD + col] = acc[r] * dinv[row];
            }
        }
    }
}

// ---------------------------------------------------------------------------
// Edge scatter: acc[dst] += msg[src]  (20 consecutive lanes = one edge's
// contiguous feature row -> coalesced gather and coalesced atomics, L2-resident)
// ---------------------------------------------------------------------------
__global__ void k_scatter(const float* __restrict__ msg,   // [n, 32] padded
                          const int* __restrict__ src, const int* __restrict__ dst,
                          float* __restrict__ acc,          // [n, 20]
                          int e) {
    int idx = blockIdx.x * blockDim.x + threadIdx.x;
    if (idx >= e * FDIM) return;                    // e*20 = 64M fits int
    int ed = idx / FDIM;
    int f = idx - ed * FDIM;
    atomicAdd(&acc[dst[ed] * FDIM + f], msg[src[ed] * MPAD + f]);
}

// ---------------------------------------------------------------------------
// Finalize: h = relu(dinv * (acc + msg_self) + b)   (msg term = self-loop)
// ---------------------------------------------------------------------------
__global__ void k_finalize(const float* __restrict__ acc,   // [n, 20]
                           const float* __restrict__ msg,   // [n, 32] padded
                           const float* __restrict__ dinv, const float* __restrict__ b,
                           float* __restrict__ h,           // [n, 20]
                           int n) {
    int idx = blockIdx.x * blockDim.x + threadIdx.x;
    if (idx >= n * FDIM) return;
    int i = idx / FDIM;
    int f = idx - i * FDIM;
    float v = fmaf(dinv[i], acc[idx] + msg[i * MPAD + f], b[f]);
    h[idx] = fmaxf(v, 0.0f);
}

// ---------------------------------------------------------------------------
// Head: out = relu(h @ fc2_w + fc2_b) @ fc3_w + fc3_b   (fused per node)
// ---------------------------------------------------------------------------
__global__ void k_head(const float* __restrict__ h, const float* __restrict__ w2,
                       const float* __restrict__ b2, const float* __restrict__ w3,
                       const float* __restrict__ b3, float* __restrict__ out, int n) {
    int i = blockIdx.x * blockDim.x + threadIdx.x;
    if (i >= n) return;
    float hv[FDIM];
#pragma unroll
    for (int k = 0; k < FDIM; ++k) hv[k] = h[i * FDIM + k];
    float o = b3[0];
#pragma unroll 4
    for (int j = 0; j < 32; ++j) {
        float s = b2[j];
#pragma unroll
        for (int k = 0; k < FDIM; ++k) s = fmaf(hv[k], w2[k * 32 + j], s);
        o = fmaf(fmaxf(s, 0.0f), w3[j], o);
    }
    out[i] = o;
}

// ---------------------------------------------------------------------------
extern "C" void kernel_launch(void* const* d_in, const int* in_sizes, int n_in,
                              void* d_out, int out_size, void* d_ws, size_t ws_size,
                              hipStream_t stream) {
    const float* x     = (const float*)d_in[0];
    const int*   ei    = (const int*)d_in[1];      // int32 (JAX default demotes int64)
    const float* fc1_w = (const float*)d_in[2];
    const float* fc1_b = (const float*)d_in[3];
    const float* Ws[3] = {(const float*)d_in[4], (const float*)d_in[6], (const float*)d_in[8]};
    const float* Bs[3] = {(const float*)d_in[5], (const float*)d_in[7], (const float*)d_in[9]};
    const float* fc2_w = (const float*)d_in[10];
    const float* fc2_b = (const float*)d_in[11];
    const float* fc3_w = (const float*)d_in[12];
    const float* fc3_b = (const float*)d_in[13];
    float* out = (float*)d_out;

    const int n = in_sizes[0] / 2;                 // 100000
    const int e = in_sizes[1] / 2;                 // 3200000
    const int* src = ei;
    const int* dst = ei + e;

    // workspace: dinv[n] | bufA[n*20] | msg[n*32] | acc[n*20]  (~29.2 MB)
    float* dinv = (float*)d_ws;
    float* bufA = dinv + n;
    float* msg  = bufA + (size_t)n * FDIM;
    float* acc  = msg  + (size_t)n * MPAD;

    const int B = 256;
    auto g = [](long long w, int b) { return (int)((w + b - 1) / b); };

    // normalization: dinv = rsqrt(1 + in-degree)
    k_deg_init<<<g(n, B), B, 0, stream>>>(dinv, n);
    k_deg_scatter<<<g(e, B), B, 0, stream>>>(dst, dinv, e);
    k_rsqrt_inplace<<<g(n, B), B, 0, stream>>>(dinv, n);

    // fc1
    k_fc1<<<g((long long)n * FDIM, B), B, 0, stream>>>(x, fc1_w, fc1_b, bufA, n);

    // three GCN layers
    const int waves = (n + 15) / 16;
    const int gemm_blocks = (waves * 32 + B - 1) / B;
    for (int l = 0; l < 3; ++l) {
        k_gemm_scale<<<gemm_blocks, B, 0, stream>>>(bufA, Ws[l], dinv, msg, n);
        hipMemsetAsync(acc, 0, (size_t)n * FDIM * sizeof(float), stream);
        k_scatter<<<g((long long)e * FDIM, B), B, 0, stream>>>(msg, src, dst, acc, e);
        k_finalize<<<g((long long)n * FDIM, B), B, 0, stream>>>(acc, msg, dinv, Bs[l], bufA, n);
    }

    // fc2 -> relu -> fc3
    k_head<<<g(n, B), B, 0, stream>>>(bufA, fc2_w, fc2_b, fc3_w, fc3_b, out, n);
}